// MultiBoxLoss_33397665693732
// MI455X (gfx1250) — compile-verified
//
#include <hip/hip_runtime.h>
#include <stdint.h>

#define NB   64
#define NP   32768
#define NC   81
#define NOBJ 32
#define TILE 128
#define THRESH 0.5f
#define VAR0 0.1f
#define VAR1 0.2f

typedef __attribute__((ext_vector_type(4))) unsigned int v4u;
typedef __attribute__((ext_vector_type(8))) int          v8i;
typedef __attribute__((ext_vector_type(4))) int          v4i;

__device__ __forceinline__ unsigned long long mbx_shfl_xor_u64(unsigned long long v, int m) {
  unsigned lo = (unsigned)(v & 0xFFFFFFFFull);
  unsigned hi = (unsigned)(v >> 32);
  lo = __shfl_xor(lo, m, 32);
  hi = __shfl_xor(hi, m, 32);
  return (((unsigned long long)hi) << 32) | (unsigned long long)lo;
}

// ---------------- init: zero scratch accumulators (replay-safe) ----------------
__global__ void mbx_init_kernel(unsigned long long* __restrict__ bppk,
                                float* __restrict__ accLoc, float* __restrict__ accPce,
                                float* __restrict__ accNeg, int* __restrict__ accNp) {
  int i = blockIdx.x * 256 + threadIdx.x;
  if (i < NB * NOBJ) bppk[i] = 0ull;
  if (i < NB) { accLoc[i] = 0.f; accPce[i] = 0.f; accNeg[i] = 0.f; accNp[i] = 0; }
}

// ---------------- match: IoU, per-prior best truth, per-truth best prior ----------------
__global__ __launch_bounds__(256)
void mbx_match_kernel(const float* __restrict__ priors, const float* __restrict__ targets,
                      float* __restrict__ bto, int* __restrict__ bti,
                      unsigned long long* __restrict__ bppk) {
  int b = blockIdx.y;
  int p = blockIdx.x * 256 + threadIdx.x;
  __shared__ float tr[NOBJ * 5];
  __shared__ unsigned long long slot[NOBJ];
  if (threadIdx.x < NOBJ * 5) tr[threadIdx.x] = targets[(size_t)b * NOBJ * 5 + threadIdx.x];
  if (threadIdx.x < NOBJ)     slot[threadIdx.x] = 0ull;
  __syncthreads();

  float4 pr = ((const float4*)priors)[p];
  float bx0 = pr.x - pr.z * 0.5f, by0 = pr.y - pr.w * 0.5f;
  float bx1 = pr.x + pr.z * 0.5f, by1 = pr.y + pr.w * 0.5f;
  float area_b = (bx1 - bx0) * (by1 - by0);

  float bestv = -1.0f; int besti = 0;
  for (int t = 0; t < NOBJ; ++t) {
    float ax0 = tr[t*5+0], ay0 = tr[t*5+1], ax1 = tr[t*5+2], ay1 = tr[t*5+3];
    float lx = fmaxf(ax0, bx0), ly = fmaxf(ay0, by0);
    float rx = fminf(ax1, bx1), ry = fminf(ay1, by1);
    float w = fmaxf(rx - lx, 0.f), h = fmaxf(ry - ly, 0.f);
    float inter = w * h;
    float area_a = (ax1 - ax0) * (ay1 - ay0);
    float ov = inter / (area_a + area_b - inter);
    if (ov > bestv) { bestv = ov; besti = t; }   // first-index tie-break (axis=0 argmax)
    // per-truth best prior: pack (iou_bits, ~p) so ties pick the LOWEST prior index
    unsigned long long pk = (((unsigned long long)__float_as_uint(ov)) << 32)
                          | (unsigned long long)(0xFFFFFFFFu - (unsigned)p);
    #pragma unroll
    for (int m = 16; m >= 1; m >>= 1) {
      unsigned long long o = mbx_shfl_xor_u64(pk, m);
      if (o > pk) pk = o;
    }
    if ((threadIdx.x & 31) == 0) atomicMax(&slot[t], pk);
  }
  bto[(size_t)b * NP + p] = bestv;
  bti[(size_t)b * NP + p] = besti;
  __syncthreads();
  if (threadIdx.x < NOBJ) atomicMax(&bppk[b * NOBJ + threadIdx.x], slot[threadIdx.x]);
}

// ---------------- override: scatter 2.0 / t at each truth's best prior ----------------
__global__ void mbx_override_kernel(const unsigned long long* __restrict__ bppk,
                                    float* __restrict__ bto, int* __restrict__ bti) {
  int b = threadIdx.x;   // 64 threads, sequential t => scatter last-wins
  for (int t = 0; t < NOBJ; ++t) {
    unsigned p = 0xFFFFFFFFu - (unsigned)(bppk[b * NOBJ + t] & 0xFFFFFFFFull);
    bto[(size_t)b * NP + p] = 2.0f;
    bti[(size_t)b * NP + p] = t;
  }
}

// ---------------- main streaming kernel: TDM tile -> LDS, CE + loc loss ----------------
__global__ __launch_bounds__(TILE)
void mbx_ce_loss_kernel(const float* __restrict__ lp, const float* __restrict__ cp,
                        const float* __restrict__ priors, const float* __restrict__ targets,
                        const float* __restrict__ bto, const int* __restrict__ bti,
                        float* __restrict__ lossc,
                        float* __restrict__ accLoc, float* __restrict__ accPce,
                        int* __restrict__ accNp) {
  int b  = blockIdx.y;
  int p0 = blockIdx.x * TILE;
  int p  = p0 + threadIdx.x;
  __shared__ float tile[TILE * NC];       // 128 x 81 f32 = 41472 B
  __shared__ float aLoc, aPce; __shared__ int aNp;
  if (threadIdx.x == 0) { aLoc = 0.f; aPce = 0.f; aNp = 0; }

  const float* row;
#if defined(__has_builtin) && __has_builtin(__builtin_amdgcn_tensor_load_to_lds)
  // Tensor DMA descriptor (CDNA5 ISA): 2D tile, rows of 81 f32.
  unsigned long long gaddr = (unsigned long long)(uintptr_t)(cp + ((size_t)b * NP + p0) * NC);
  unsigned ldsOff = (unsigned)(uintptr_t)&tile[0];   // flat LDS aperture: low 32 bits = LDS byte offset
  v4u g0 = { 1u,                                                      // count=1, user mode
             ldsOff,                                                  // lds_addr [63:32]
             (unsigned)(gaddr & 0xFFFFFFFFull),                       // global_addr lo
             (unsigned)((gaddr >> 32) & 0x1FFFFFFull) | (2u << 30) }; // global_addr hi | type=2
  v8i g1 = { (int)(2u << 16),                 // data_size = 4B (code 2)
             (int)(81u << 16),                // tensor_dim0 = 81 (bits 63:48)
             0,                               // tensor_dim0 hi / tensor_dim1 lo16 (TD1 = 1<<20)
             (int)(16u | (81u << 16)),        // tensor_dim1 hi16 = 16 | tile_dim0 = 81
             (int)TILE,                       // tile_dim1 = 128 rows, tile_dim2 = 0
             81, 0, 0 };                      // tensor_dim0_stride = 81 elements
  v4i gz4 = { 0, 0, 0, 0 };                   // groups 2/3 unused for 2D tiles
  v8i gz8 = { 0, 0, 0, 0, 0, 0, 0, 0 };       // extra group (6-arg toolchain form)
  if (threadIdx.x < 32)                       // one wave issues the DMA (EXEC ignored by TDM)
    __builtin_amdgcn_tensor_load_to_lds(g0, g1, gz4, gz4, gz8, 0);
  __builtin_amdgcn_s_wait_tensorcnt(0);
  __syncthreads();
  row = &tile[threadIdx.x * NC];              // 81 dwords/row: conflict-free vs 64 banks
#else
  __syncthreads();
  row = cp + ((size_t)b * NP + p) * NC;
#endif

  // log-softmax over 81 classes
  float m = -1e30f;
  #pragma unroll 9
  for (int c = 0; c < NC; ++c) m = fmaxf(m, row[c]);
  float s = 0.f;
  #pragma unroll 9
  for (int c = 0; c < NC; ++c) s += __expf(row[c] - m);
  float lse = m + __logf(s);

  size_t bp = (size_t)b * NP + p;
  float ov = bto[bp];
  int   t  = bti[bp];
  const float* tb = &targets[((size_t)b * NOBJ + t) * 5];
  int lab = (int)tb[4];
  int cls = (ov < THRESH) ? 0 : (lab + 1);
  bool pos = cls > 0;
  float ce = lse - row[cls];
  lossc[bp] = pos ? 0.f : ce;

  // encode + smooth-L1 (positives only)
  float4 pr = ((const float4*)priors)[p];
  float mcx = (tb[0] + tb[2]) * 0.5f, mcy = (tb[1] + tb[3]) * 0.5f;
  float mw  = tb[2] - tb[0],          mh  = tb[3] - tb[1];
  float l0 = (mcx - pr.x) / (VAR0 * pr.z);
  float l1 = (mcy - pr.y) / (VAR0 * pr.w);
  float l2 = __logf(mw / pr.z) / VAR1;
  float l3 = __logf(mh / pr.w) / VAR1;
  float4 lpv = ((const float4*)lp)[bp];
  float ll = 0.f;
  { float d = lpv.x - l0, a = fabsf(d); ll += (a < 1.f) ? 0.5f*d*d : a - 0.5f; }
  { float d = lpv.y - l1, a = fabsf(d); ll += (a < 1.f) ? 0.5f*d*d : a - 0.5f; }
  { float d = lpv.z - l2, a = fabsf(d); ll += (a < 1.f) ? 0.5f*d*d : a - 0.5f; }
  { float d = lpv.w - l3, a = fabsf(d); ll += (a < 1.f) ? 0.5f*d*d : a - 0.5f; }

  atomicAdd(&aLoc, pos ? ll : 0.f);
  atomicAdd(&aPce, pos ? ce : 0.f);
  if (pos) atomicAdd(&aNp, 1);
  __syncthreads();
  if (threadIdx.x == 0) {
    atomicAdd(&accLoc[b], aLoc);
    atomicAdd(&accPce[b], aPce);
    atomicAdd(&accNp[b], aNp);
  }
}

// ---------------- hard-negative mining: exact top-k sum via bit bisection ----------------
__global__ __launch_bounds__(1024)
void mbx_topk_kernel(const float* __restrict__ lossc, const int* __restrict__ accNp,
                     float* __restrict__ accNeg) {
  int b = blockIdx.x;
  const float* src = lossc + (size_t)b * NP;
  unsigned v[32];
  #pragma unroll
  for (int i = 0; i < 32; ++i) v[i] = __float_as_uint(src[i * 1024 + threadIdx.x]);
  int k = min(3 * accNp[b], NP - 1);

  __shared__ unsigned scnt; __shared__ float ssum; __shared__ unsigned scg;
  unsigned lo = 0u, hi = 0x7F800001u;   // loss_c >= 0: IEEE bits monotone as u32
  if (k > 0) {
    while (hi - lo > 1u) {
      unsigned mid = lo + (hi - lo) / 2u;
      if (threadIdx.x == 0) scnt = 0u;
      __syncthreads();
      unsigned c = 0;
      #pragma unroll
      for (int i = 0; i < 32; ++i) c += (v[i] >= mid) ? 1u : 0u;
      atomicAdd(&scnt, c);
      __syncthreads();
      if (scnt >= (unsigned)k) lo = mid; else hi = mid;
      __syncthreads();
    }
  }
  if (threadIdx.x == 0) { ssum = 0.f; scg = 0u; }
  __syncthreads();
  float s = 0.f; unsigned cg = 0;
  #pragma unroll
  for (int i = 0; i < 32; ++i)
    if (v[i] > lo) { s += __uint_as_float(v[i]); ++cg; }
  atomicAdd(&ssum, s); atomicAdd(&scg, cg);
  __syncthreads();
  if (threadIdx.x == 0) {
    float topk = 0.f;
    if (k > 0) topk = ssum + (float)(k - (int)scg) * __uint_as_float(lo);
    accNeg[b] = topk;
  }
}

// ---------------- final scalar ----------------
__global__ void mbx_final_kernel(const float* __restrict__ accLoc, const float* __restrict__ accPce,
                                 const float* __restrict__ accNeg, const int* __restrict__ accNp,
                                 float* __restrict__ out) {
  __shared__ float sl[NB], sc[NB]; __shared__ int sn[NB];
  int i = threadIdx.x;
  sl[i] = accLoc[i]; sc[i] = accPce[i] + accNeg[i]; sn[i] = accNp[i];
  __syncthreads();
  if (i == 0) {
    float L = 0.f, C = 0.f; int N = 0;
    for (int j = 0; j < NB; ++j) { L += sl[j]; C += sc[j]; N += sn[j]; }
    out[0] = (L + C) / (float)N;
  }
}

extern "C" void kernel_launch(void* const* d_in, const int* in_sizes, int n_in,
                              void* d_out, int out_size, void* d_ws, size_t ws_size,
                              hipStream_t stream) {
  (void)in_sizes; (void)n_in; (void)out_size; (void)ws_size;
  const float* lp      = (const float*)d_in[0];   // loc_preds  (B,P,4)
  const float* cp      = (const float*)d_in[1];   // conf_preds (B,P,81)
  const float* priors  = (const float*)d_in[2];   // (P,4)
  const float* targets = (const float*)d_in[3];   // (B,32,5)

  char* ws = (char*)d_ws;
  const size_t BP = (size_t)NB * NP;
  float*              bto    = (float*)(ws);                    // B*P f32
  int*                bti    = (int*)  (ws + BP * 4);           // B*P i32
  float*              lossc  = (float*)(ws + BP * 8);           // B*P f32
  unsigned long long* bppk   = (unsigned long long*)(ws + BP * 12);
  float*              accLoc = (float*)(ws + BP * 12 + (size_t)NB * NOBJ * 8);
  float*              accPce = accLoc + NB;
  float*              accNeg = accPce + NB;
  int*                accNp  = (int*)(accNeg + NB);

  mbx_init_kernel<<<(NB * NOBJ + 255) / 256, 256, 0, stream>>>(bppk, accLoc, accPce, accNeg, accNp);
  mbx_match_kernel<<<dim3(NP / 256, NB), 256, 0, stream>>>(priors, targets, bto, bti, bppk);
  mbx_override_kernel<<<1, NB, 0, stream>>>(bppk, bto, bti);
  mbx_ce_loss_kernel<<<dim3(NP / TILE, NB), TILE, 0, stream>>>(lp, cp, priors, targets, bto, bti,
                                                               lossc, accLoc, accPce, accNp);
  mbx_topk_kernel<<<NB, 1024, 0, stream>>>(lossc, accNp, accNeg);
  mbx_final_kernel<<<1, NB, 0, stream>>>(accLoc, accPce, accNeg, accNp, (float*)d_out);
}